// GAT_48919677501958
// MI455X (gfx1250) — compile-verified
//
#include <hip/hip_runtime.h>
#include <hip/hip_bf16.h>
#include <math.h>

typedef float v2f __attribute__((ext_vector_type(2)));
typedef float v8f __attribute__((ext_vector_type(8)));

#define NEG_SLOPE 0.2f

__device__ __forceinline__ float leaky(float v) {
    return v > 0.0f ? v : NEG_SLOPE * v;
}

// Float atomic-max via int bit tricks (works for mixed signs, init = -inf).
__device__ __forceinline__ void atomicMaxF32(float* addr, float val) {
    if (val >= 0.0f) atomicMax((int*)addr, __float_as_int(val));
    else             atomicMin((unsigned int*)addr, __float_as_uint(val));
}

// ---------------------------------------------------------------------------
// GEMM C[M x ldc] = A[M x K] @ B[K x N] via V_WMMA_F32_16X16X4_F32.
// Requirements: K % 8 == 0, N % 16 == 0 (B pre-padded), ldc >= N.
// One wave (32 lanes) computes a 16-row x (NT*16)-col strip.
//
// Pipeline per 8-wide K chunk:
//   - A tile (16x8) prefetched one chunk ahead (global_load_b128, clamped addr)
//   - staged in LDS (row stride 12: 16B-aligned float4 stores, conflict-free
//     b64 fragment reads: 12*r mod 64 distinct for r<16)
//   - all B fragments for the chunk batch-loaded into registers first,
//     then 2 x NT WMMAs run back-to-back.
//
// WMMA fragment layouts (CDNA5 ISA):
//   A 16x4 : lanes 0-15 -> K = {k, k+1}; lanes 16-31 -> K = {k+2, k+3}
//   B 4x16 : same K split per lane-half, N = lane % 16
//   C 16x16: VGPR r -> row half*8 + r, col = lane % 16
// ---------------------------------------------------------------------------
template<int NT>
__global__ void __launch_bounds__(32)
gemm_wmma_f32(const float* __restrict__ A, const float* __restrict__ B,
              float* __restrict__ C, int M, int K, int N, int ldc)
{
    __shared__ float As[16 * 12];       // 768 B

    const int lane = (int)threadIdx.x;
    const int half = lane >> 4;          // 0 or 1
    const int l16  = lane & 15;

    // Cooperative A-chunk load: lane -> (row, 4-col group)
    const int alr = lane >> 1;           // 0..15
    const int alc = (lane & 1) * 4;      // 0 or 4
    int rowA = blockIdx.x * 16 + alr;
    if (rowA >= M) rowA = M - 1;         // clamp loads; garbage masked at store
    const float* __restrict__ Aload = A + (size_t)rowA * (size_t)K + alc;

    const int ncol0 = blockIdx.y * (16 * NT);
    const int colB  = ncol0 + l16;       // per-lane base column (tile t adds t*16)

    v8f acc[NT];
#pragma unroll
    for (int t = 0; t < NT; ++t) {
        v8f z = {0.f, 0.f, 0.f, 0.f, 0.f, 0.f, 0.f, 0.f};
        acc[t] = z;
    }

    float4 av = *(const float4*)(Aload);            // prologue A prefetch

    for (int kc = 0; kc < K; kc += 8) {
        __syncthreads();                            // LDS reuse fence
        *(float4*)(&As[alr * 12 + alc]) = av;

        // Batch-prefetch all B fragments for this chunk (2 k-steps x NT tiles).
        v2f breg[2][NT];
#pragma unroll
        for (int kk2 = 0; kk2 < 2; ++kk2) {
            const float* __restrict__ Bk =
                B + (size_t)(kc + kk2 * 4 + 2 * half) * (size_t)N + colB;
#pragma unroll
            for (int t = 0; t < NT; ++t) {
                breg[kk2][t].x = Bk[t * 16];
                breg[kk2][t].y = Bk[N + t * 16];
            }
        }

        // Prefetch next chunk's A tile (uniform clamped address: no branch).
        const int kn = (kc + 8 < K) ? (kc + 8) : 0;
        av = *(const float4*)(Aload + kn);

        __syncthreads();                            // LDS stores visible

#pragma unroll
        for (int kk2 = 0; kk2 < 2; ++kk2) {
            const v2f a = *(const v2f*)(&As[l16 * 12 + kk2 * 4 + 2 * half]);
#pragma unroll
            for (int t = 0; t < NT; ++t) {
                acc[t] = __builtin_amdgcn_wmma_f32_16x16x4_f32(
                    false, a, false, breg[kk2][t], (short)0, acc[t], false, false);
            }
        }
    }

    // Epilogue: wave-uniform bounds check -> unguarded store clause on the
    // fast path (full tiles); divergent guards only for a ragged last tile.
    const int crow0 = blockIdx.x * 16 + half * 8;
    float* __restrict__ Cbase = C + (size_t)crow0 * (size_t)ldc + ncol0 + l16;
    if (blockIdx.x * 16 + 16 <= M) {
#pragma unroll
        for (int t = 0; t < NT; ++t) {
            float* __restrict__ Cp = Cbase + t * 16;
#pragma unroll
            for (int r = 0; r < 8; ++r)
                Cp[(size_t)r * (size_t)ldc] = acc[t][r];
        }
    } else {
#pragma unroll
        for (int t = 0; t < NT; ++t) {
            float* __restrict__ Cp = Cbase + t * 16;
#pragma unroll
            for (int r = 0; r < 8; ++r)
                if (crow0 + r < M) Cp[(size_t)r * (size_t)ldc] = acc[t][r];
        }
    }
}

// ---------------------------------------------------------------------------
__global__ void fill_f32(float* __restrict__ p, int n, float v) {
    int t = blockIdx.x * blockDim.x + threadIdx.x;
    if (t < n) p[t] = v;
}

// Zero-pad W2 [K x N] -> W2p [K x Np]
__global__ void pad_cols(const float* __restrict__ src, float* __restrict__ dst,
                         int K, int N, int Np)
{
    int t = blockIdx.x * blockDim.x + threadIdx.x;
    if (t >= K * Np) return;
    int r = t / Np, c = t - r * Np;
    dst[t] = (c < N) ? src[r * N + c] : 0.f;
}

// Layer 1: per-(node,head) attention halves. h1 is [N, 8*64] row-major.
__global__ void alpha1_kernel(const float* __restrict__ h1,
                              const float* __restrict__ a_src,
                              const float* __restrict__ a_dst,
                              float* __restrict__ as, float* __restrict__ ad, int Nn)
{
    int t = blockIdx.x * blockDim.x + threadIdx.x;
    if (t >= Nn * 8) return;
    int n = t >> 3, h = t & 7;
    const float* hp = h1 + (size_t)n * 512 + h * 64;
    const float* sp = a_src + h * 64;
    const float* dp = a_dst + h * 64;
    float ss = 0.f, dd = 0.f;
#pragma unroll 8
    for (int c = 0; c < 64; ++c) { float v = hp[c]; ss += v * sp[c]; dd += v * dp[c]; }
    as[t] = ss;
    ad[t] = dd;
}

__device__ __forceinline__ void edge_sd(const int* __restrict__ ei, int E, int e,
                                        int& s, int& d)
{
    if (e < E) { s = ei[e]; d = ei[E + e]; }
    else       { s = e - E; d = e - E; }     // self loops appended
}

__global__ void edge_max1(const int* __restrict__ ei, int E, int Nn,
                          const float* __restrict__ as, const float* __restrict__ ad,
                          float* __restrict__ mx)
{
    int t = blockIdx.x * blockDim.x + threadIdx.x;
    if (t >= (E + Nn) * 8) return;
    int e = t >> 3, h = t & 7;
    int s, d; edge_sd(ei, E, e, s, d);
    atomicMaxF32(&mx[d * 8 + h], leaky(as[s * 8 + h] + ad[d * 8 + h]));
}

__global__ void edge_expsum1(const int* __restrict__ ei, int E, int Nn,
                             const float* __restrict__ as, const float* __restrict__ ad,
                             const float* __restrict__ mx,
                             float* __restrict__ ex, float* __restrict__ denom)
{
    int t = blockIdx.x * blockDim.x + threadIdx.x;
    if (t >= (E + Nn) * 8) return;
    int e = t >> 3, h = t & 7;
    int s, d; edge_sd(ei, E, e, s, d);
    float v  = leaky(as[s * 8 + h] + ad[d * 8 + h]);
    float xv = expf(v - mx[d * 8 + h]);
    ex[t] = xv;
    atomicAdd(&denom[d * 8 + h], xv);
}

// Weighted scatter-add: one thread per (edge, 4-channel group); 128 groups/edge.
__global__ void aggregate1(const int* __restrict__ ei, int E, int Nn,
                           const float* __restrict__ ex, const float* __restrict__ denom,
                           const float* __restrict__ h1, float* __restrict__ out)
{
    int t = blockIdx.x * blockDim.x + threadIdx.x;
    if (t >= (E + Nn) * 128) return;
    int e = t >> 7;
    int g = t & 127;
    int h = g >> 4;
    int s, d; edge_sd(ei, E, e, s, d);
    float alpha = ex[e * 8 + h] / denom[d * 8 + h];
    const float4 hv = *(const float4*)(h1 + (size_t)s * 512 + g * 4);
    float* o = out + (size_t)d * 512 + g * 4;
    atomicAdd(o + 0, hv.x * alpha);
    atomicAdd(o + 1, hv.y * alpha);
    atomicAdd(o + 2, hv.z * alpha);
    atomicAdd(o + 3, hv.w * alpha);
}

__global__ void bias_elu(float* __restrict__ h2, const float* __restrict__ b, int total) {
    int t = blockIdx.x * blockDim.x + threadIdx.x;
    if (t >= total) return;
    float v = h2[t] + b[t & 511];
    h2[t] = v > 0.f ? v : expm1f(v);
}

// Layer 2 (1 head, 40 channels, h3 padded to ldc=48)
__global__ void alpha2_kernel(const float* __restrict__ h3,
                              const float* __restrict__ a_src,
                              const float* __restrict__ a_dst,
                              float* __restrict__ as, float* __restrict__ ad, int Nn)
{
    int n = blockIdx.x * blockDim.x + threadIdx.x;
    if (n >= Nn) return;
    const float* hp = h3 + (size_t)n * 48;
    float ss = 0.f, dd = 0.f;
#pragma unroll 8
    for (int c = 0; c < 40; ++c) { float v = hp[c]; ss += v * a_src[c]; dd += v * a_dst[c]; }
    as[n] = ss;
    ad[n] = dd;
}

__global__ void edge_max2(const int* __restrict__ ei, int E, int Nn,
                          const float* __restrict__ as, const float* __restrict__ ad,
                          float* __restrict__ mx)
{
    int e = blockIdx.x * blockDim.x + threadIdx.x;
    if (e >= E + Nn) return;
    int s, d; edge_sd(ei, E, e, s, d);
    atomicMaxF32(&mx[d], leaky(as[s] + ad[d]));
}

__global__ void edge_expsum2(const int* __restrict__ ei, int E, int Nn,
                             const float* __restrict__ as, const float* __restrict__ ad,
                             const float* __restrict__ mx,
                             float* __restrict__ ex, float* __restrict__ denom)
{
    int e = blockIdx.x * blockDim.x + threadIdx.x;
    if (e >= E + Nn) return;
    int s, d; edge_sd(ei, E, e, s, d);
    float v  = leaky(as[s] + ad[d]);
    float xv = expf(v - mx[d]);
    ex[e] = xv;
    atomicAdd(&denom[d], xv);
}

// One thread per (edge, 4-channel group); 10 groups cover 40 channels.
__global__ void aggregate2(const int* __restrict__ ei, int E, int Nn,
                           const float* __restrict__ ex, const float* __restrict__ denom,
                           const float* __restrict__ h3, float* __restrict__ out)
{
    int t = blockIdx.x * blockDim.x + threadIdx.x;
    if (t >= (E + Nn) * 10) return;
    int e = t / 10;
    int g = t - e * 10;
    int s, d; edge_sd(ei, E, e, s, d);
    float alpha = ex[e] / denom[d];
    const float4 hv = *(const float4*)(h3 + (size_t)s * 48 + g * 4);
    float* o = out + (size_t)d * 40 + g * 4;
    atomicAdd(o + 0, hv.x * alpha);
    atomicAdd(o + 1, hv.y * alpha);
    atomicAdd(o + 2, hv.z * alpha);
    atomicAdd(o + 3, hv.w * alpha);
}

__global__ void bias2_kernel(float* __restrict__ out, const float* __restrict__ b, int total) {
    int t = blockIdx.x * blockDim.x + threadIdx.x;
    if (t >= total) return;
    out[t] += b[t % 40];
}

// ---------------------------------------------------------------------------
extern "C" void kernel_launch(void* const* d_in, const int* in_sizes, int n_in,
                              void* d_out, int out_size, void* d_ws, size_t ws_size,
                              hipStream_t stream)
{
    const float* x    = (const float*)d_in[0];
    const int*   ei   = (const int*)  d_in[1];
    const float* W1   = (const float*)d_in[2];
    const float* aS1  = (const float*)d_in[3];
    const float* aD1  = (const float*)d_in[4];
    const float* b1   = (const float*)d_in[5];
    const float* W2   = (const float*)d_in[6];
    const float* aS2  = (const float*)d_in[7];
    const float* aD2  = (const float*)d_in[8];
    const float* b2   = (const float*)d_in[9];
    float* out = (float*)d_out;

    const int IN  = 256, F1 = 512, OUT = 40, OUTP = 48;
    const int Nn  = in_sizes[0] / IN;        // 10000
    const int E   = in_sizes[1] / 2;         // 160000
    const int ET  = E + Nn;                  // +self loops

    // Workspace carve-up (floats)
    float* ws  = (float*)d_ws;
    float* h1  = ws;                          // [N, 512]
    float* h2  = h1  + (size_t)Nn * F1;       // [N, 512] (layer-1 out / layer-2 in)
    float* as1 = h2  + (size_t)Nn * F1;       // [N, 8]
    float* ad1 = as1 + (size_t)Nn * 8;
    float* mx1 = ad1 + (size_t)Nn * 8;
    float* dn1 = mx1 + (size_t)Nn * 8;
    float* ex1 = dn1 + (size_t)Nn * 8;        // [ET, 8]
    float* h3  = ex1 + (size_t)ET * 8;        // [N, 48] (40 padded)
    float* as2 = h3  + (size_t)Nn * 48;       // [N]
    float* ad2 = as2 + Nn;
    float* mx2 = ad2 + Nn;
    float* dn2 = mx2 + Nn;
    float* ex2 = dn2 + Nn;                    // [ET]
    float* W2p = ex2 + ET;                    // [512, 48] zero-padded W2

    const int B = 256;
    const float NEG_INF = -__builtin_inff();

    // ---------------- Layer 1 ----------------
    // h1 = x @ W1  (M=10000, K=256, N=512): 625 x 8 strips of 16x64
    gemm_wmma_f32<4><<<dim3((Nn + 15) / 16, F1 / 64), 32, 0, stream>>>(
        x, W1, h1, Nn, IN, F1, F1);

    alpha1_kernel<<<(Nn * 8 + B - 1) / B, B, 0, stream>>>(h1, aS1, aD1, as1, ad1, Nn);

    fill_f32<<<(Nn * 8 + B - 1) / B, B, 0, stream>>>(mx1, Nn * 8, NEG_INF);
    fill_f32<<<(Nn * 8 + B - 1) / B, B, 0, stream>>>(dn1, Nn * 8, 0.f);
    fill_f32<<<(Nn * F1 + B - 1) / B, B, 0, stream>>>(h2, Nn * F1, 0.f);

    edge_max1   <<<(ET * 8 + B - 1) / B, B, 0, stream>>>(ei, E, Nn, as1, ad1, mx1);
    edge_expsum1<<<(ET * 8 + B - 1) / B, B, 0, stream>>>(ei, E, Nn, as1, ad1, mx1, ex1, dn1);
    aggregate1  <<<(ET * 128 + B - 1) / B, B, 0, stream>>>(ei, E, Nn, ex1, dn1, h1, h2);

    bias_elu<<<(Nn * F1 + B - 1) / B, B, 0, stream>>>(h2, b1, Nn * F1);

    // ---------------- Layer 2 ----------------
    // Pad W2 to [512 x 48] so the GEMM inner loop is guard-free.
    pad_cols<<<(F1 * OUTP + B - 1) / B, B, 0, stream>>>(W2, W2p, F1, OUT, OUTP);

    // h3 = h2 @ W2p  (M=10000, K=512, N=48): 625 x 1 strips of 16x48
    gemm_wmma_f32<3><<<dim3((Nn + 15) / 16, 1), 32, 0, stream>>>(
        h2, W2p, h3, Nn, F1, OUTP, OUTP);

    alpha2_kernel<<<(Nn + B - 1) / B, B, 0, stream>>>(h3, aS2, aD2, as2, ad2, Nn);

    fill_f32<<<(Nn + B - 1) / B, B, 0, stream>>>(mx2, Nn, NEG_INF);
    fill_f32<<<(Nn + B - 1) / B, B, 0, stream>>>(dn2, Nn, 0.f);
    fill_f32<<<(Nn * OUT + B - 1) / B, B, 0, stream>>>(out, Nn * OUT, 0.f);

    edge_max2   <<<(ET + B - 1) / B, B, 0, stream>>>(ei, E, Nn, as2, ad2, mx2);
    edge_expsum2<<<(ET + B - 1) / B, B, 0, stream>>>(ei, E, Nn, as2, ad2, mx2, ex2, dn2);
    aggregate2  <<<(ET * 10 + B - 1) / B, B, 0, stream>>>(ei, E, Nn, ex2, dn2, h3, out);

    bias2_kernel<<<(Nn * OUT + B - 1) / B, B, 0, stream>>>(out, b2, Nn * OUT);
}